// Mamba_85555748536904
// MI455X (gfx1250) — compile-verified
//
#include <hip/hip_runtime.h>
#include <math.h>

// ---------------------------------------------------------------------------
// Mamba block forward for MI455X (gfx1250, wave32, WMMA, TDM).
// fp32 reference; whole layer (~27 GFLOP, ~80 MB) is L2-resident (192 MB) ->
// bandwidth/latency bound; use V_WMMA_F32_16X16X4_F32 (exact fp32) for all
// projections and the Tensor Data Mover for the scan's B/C staging.
// GEMM inner loop is software-pipelined 1 step deep so the WMMA group never
// waits on the loads issued in the same step (one partial s_wait per 8 WMMAs
// instead of a blocking wait per WMMA).
// ---------------------------------------------------------------------------

typedef float v2f __attribute__((ext_vector_type(2)));
typedef float v8f __attribute__((ext_vector_type(8)));
typedef unsigned int u32x4 __attribute__((ext_vector_type(4)));
typedef unsigned int u32x8 __attribute__((ext_vector_type(8)));

#define D_MODEL 1024
#define D_STATE 16
#define D_CONV  4
#define D_INNER 2048
#define DT_RANK 64
#define BB 2
#define LL 1024
#define NTOK (BB * LL)   // 2048 tokens

// ---------------------------------------------------------------------------
// RMSNorm (norm/sqrt(D) variant): one block per token, 256 threads x float4.
// ---------------------------------------------------------------------------
__global__ void __launch_bounds__(256, 1)
rmsnorm_kernel(const float* __restrict__ h,
               const float* __restrict__ scale,
               float* __restrict__ hn)
{
    int tok = blockIdx.x;
    int tid = threadIdx.x;
    const float4* hrow = (const float4*)(h + (size_t)tok * D_MODEL);
    float4 v = hrow[tid];
    float ss = v.x * v.x + v.y * v.y + v.z * v.z + v.w * v.w;
    for (int off = 16; off > 0; off >>= 1)        // wave32 reduce
        ss += __shfl_down(ss, off, 32);
    __shared__ float red[8];
    int wid = tid >> 5, lane = tid & 31;
    if (lane == 0) red[wid] = ss;
    __syncthreads();
    float tot = red[0] + red[1] + red[2] + red[3] +
                red[4] + red[5] + red[6] + red[7];
    float norm = sqrtf(tot / (float)D_MODEL);
    float inv  = 1.0f / (norm + 1e-8f);
    float4 s4 = ((const float4*)scale)[tid];
    float4 o;
    o.x = v.x * s4.x * inv;
    o.y = v.y * s4.y * inv;
    o.z = v.z * s4.z * inv;
    o.w = v.w * s4.w * inv;
    ((float4*)(hn + (size_t)tok * D_MODEL))[tid] = o;
}

// ---------------------------------------------------------------------------
// fp32 WMMA GEMM:  C[M,N] = A[M,K(lda)] * W[N,K]^T  (+ epilogue)
//   NT   : 16-wide N sub-tiles per wave (wave tile = 16M x 16*NT N)
//   WN   : waves in N direction (WM = 8/WN waves in M); block = 256 thr
//   MODE : 0 plain | 1 softplus(acc+aux[col]) | 2 acc+aux[row*ldaux+col]
// All geometry compile-time -> branch-free inner loop, EXEC stays all-1s.
// Software pipeline: fragments for step k+4 are loaded before the WMMAs for
// step k execute, so each step's WMMA group only needs loadcnt <= (in-flight
// next-step loads) rather than a full wait per WMMA.
// ---------------------------------------------------------------------------
template <int NT, int WN, int MODE>
__global__ void __launch_bounds__(256, 1)
gemm_wmma_f32(const float* __restrict__ A, int lda,
              const float* __restrict__ W, int K,
              float* __restrict__ C, int ldc,
              const float* __restrict__ aux, int ldaux)
{
    constexpr int WM = 8 / WN;               // waves in M
    int tid  = threadIdx.x;
    int wave = tid >> 5;
    int lane = tid & 31;
    int wm = wave % WM;
    int wn = wave / WM;
    int m0    = blockIdx.x * (16 * WM) + wm * 16;
    int nbase = blockIdx.y * (16 * NT * WN) + wn * (16 * NT);
    int lr = lane & 15;
    int lh = lane >> 4;                      // 0 or 1

    const float* arow = A + (size_t)(m0 + lr) * lda + 2 * lh;
    const float* wp[NT];
#pragma unroll
    for (int t = 0; t < NT; t++)
        wp[t] = W + (size_t)(nbase + 16 * t + lr) * K + 2 * lh;

    v8f acc[NT];
#pragma unroll
    for (int t = 0; t < NT; t++) acc[t] = (v8f){};

    // pipeline prologue: fragments for k = 0
    v2f a_cur = *(const v2f*)(arow);
    v2f b_cur[NT];
#pragma unroll
    for (int t = 0; t < NT; t++) b_cur[t] = *(const v2f*)(wp[t]);

    for (int k0 = 0; k0 < K; k0 += 64) {
        // keep the streams warm (global_prefetch_b8); speculative past end OK
        __builtin_prefetch(arow + k0 + 128, 0, 3);
        __builtin_prefetch(wp[0] + k0 + 128, 0, 3);
#pragma unroll
        for (int kk = 0; kk < 64; kk += 4) {
            int knext = k0 + kk + 4;
            v2f a_nxt = a_cur;               // defined in the final-step path
            v2f b_nxt[NT];
#pragma unroll
            for (int t = 0; t < NT; t++) b_nxt[t] = b_cur[t];
            if (kk < 60) {                   // compile-time: stay inside chunk
                a_nxt = *(const v2f*)(arow + knext);
#pragma unroll
                for (int t = 0; t < NT; t++)
                    b_nxt[t] = *(const v2f*)(wp[t] + knext);
            } else if (knext < K) {          // wave-uniform, once per 16 steps
                a_nxt = *(const v2f*)(arow + knext);
#pragma unroll
                for (int t = 0; t < NT; t++)
                    b_nxt[t] = *(const v2f*)(wp[t] + knext);
            }
#pragma unroll
            for (int t = 0; t < NT; t++)     // consume PREVIOUS step's frags
                acc[t] = __builtin_amdgcn_wmma_f32_16x16x4_f32(
                    false, a_cur, false, b_cur[t], (short)0, acc[t], false, false);
            a_cur = a_nxt;
#pragma unroll
            for (int t = 0; t < NT; t++) b_cur[t] = b_nxt[t];
        }
    }

    int rbase = m0 + lh * 8;
#pragma unroll
    for (int t = 0; t < NT; t++) {
        int col = nbase + 16 * t + lr;
#pragma unroll
        for (int i = 0; i < 8; i++) {
            int row = rbase + i;
            float v = acc[t][i];
            if constexpr (MODE == 1) {               // bias + softplus
                v += aux[col];
                v = (v > 20.0f) ? v : log1pf(__expf(v));
            } else if constexpr (MODE == 2) {        // residual add
                v += aux[(size_t)row * ldaux + col];
            }
            C[(size_t)row * ldc + col] = v;
        }
    }
}

// ---------------------------------------------------------------------------
// Causal depthwise conv (width 4) + bias + SiLU over the x-half of xz.
// xz layout: [b, l, 4096] with x = cols [0,2048), z = cols [2048,4096).
// ---------------------------------------------------------------------------
__global__ void __launch_bounds__(256, 1)
conv_silu_kernel(const float* __restrict__ xz,
                 const float* __restrict__ w,
                 const float* __restrict__ bias,
                 float* __restrict__ xc)
{
    int idx = blockIdx.x * 256 + threadIdx.x;      // over B*L*D_INNER
    int c  = idx & (D_INNER - 1);
    int l  = (idx >> 11) & (LL - 1);
    int b  = idx >> 21;
    const float* xcol = xz + ((size_t)(b * LL) + l) * (2 * D_INNER) + c;
    const int strideL = 2 * D_INNER;
    float acc = bias[c];
    float w0 = w[c * 4 + 0], w1 = w[c * 4 + 1];
    float w2 = w[c * 4 + 2], w3 = w[c * 4 + 3];
    if (l >= 3) acc += w0 * xcol[-3 * strideL];
    if (l >= 2) acc += w1 * xcol[-2 * strideL];
    if (l >= 1) acc += w2 * xcol[-1 * strideL];
    acc += w3 * xcol[0];
    acc = acc / (1.0f + __expf(-acc));             // SiLU
    xc[idx] = acc;
}

// ---------------------------------------------------------------------------
// TDM: async-load a 2D tile (16 tokens x 32 floats, row stride 96 floats)
// from x_dbl into LDS via tensor_load_to_lds (CDNA5 ISA, D# groups 0+1,
// 2D form). TDM writes tile rows contiguously -> LDS layout [16][32].
// Issued by one wave; completion tracked with TENSORcnt (in-order per wave).
// ---------------------------------------------------------------------------
__device__ __forceinline__ void tdm_load_tile16x32(unsigned lds_byte_addr,
                                                   const float* gptr)
{
    unsigned long long ga = (unsigned long long)(size_t)gptr;
    // D# group 0: count=1 | lds_addr | global_addr[56:0] | type=2
    u32x4 g0 = { 1u,
                 lds_byte_addr,
                 (unsigned)ga,
                 (unsigned)((ga >> 32) & 0x01FFFFFFu) | (2u << 30) };
    // D# group 1: data_size=4B(code 2); tensor_dim0=32, tensor_dim1=16;
    // tile_dim0=32, tile_dim1=16; tensor_dim0_stride=96. Bit-packed per ISA:
    //  w0[17:16]=data_size  w1[31:16]=tdim0.lo  w2[31:16]=tdim1.lo
    //  w3[31:16]=tile_dim0  w4[15:0]=tile_dim1  w5=tdim0_stride.lo
    u32x8 g1 = { (2u << 16),
                 (32u << 16),
                 (16u << 16),
                 (32u << 16),
                 16u,
                 96u,
                 0u,
                 0u };
    asm volatile("tensor_load_to_lds %0, %1" : : "s"(g0), "s"(g1) : "memory");
}

// ---------------------------------------------------------------------------
// Selective scan: one lane per channel d, 16-state recurrence in registers.
// B/C vectors (32 floats/token, shared by all 2048 channels of a batch) are
// TDM double-buffered through LDS: wave 0 issues chunk i+1, waits
// s_wait_tensorcnt<=1 (chunk i done, in-order), barrier publishes, everyone
// processes chunk i while chunk i+1 streams in. Fuses +D*x and *silu(z);
// writes y in-place over the conv buffer (element read before written).
// ---------------------------------------------------------------------------
__global__ void __launch_bounds__(256, 1)
scan_kernel(const float* __restrict__ xdbl,   // [NTOK,96]
            const float* __restrict__ dtb,    // [NTOK,D_INNER]
            const float* __restrict__ xz,     // z half
            float* __restrict__ xc,           // in: conv x, out: y
            const float* __restrict__ A_log,
            const float* __restrict__ Dp)
{
    const int CH  = 16;
    const int NCH = LL / CH;                  // 64 chunks
    int b    = blockIdx.x >> 3;
    int dblk = blockIdx.x & 7;
    int tid  = threadIdx.x;
    int wave = tid >> 5;
    int d    = dblk * 256 + tid;

    float Ad[D_STATE], s[D_STATE];
#pragma unroll
    for (int n = 0; n < D_STATE; n++) {
        Ad[n] = -__expf(A_log[(size_t)d * D_STATE + n]);
        s[n]  = 0.0f;
    }
    float Dd = Dp[d];

    __shared__ float lBC[2][CH][32];          // [buf][token][B(16)|C(16)]
    unsigned lds0 = (unsigned)(size_t)&lBC[0][0][0];
    unsigned lds1 = (unsigned)(size_t)&lBC[1][0][0];
    const float* gbase = xdbl + (size_t)(b * LL) * 96 + DT_RANK;

    if (wave == 0)                            // prologue: chunk 0 -> buf 0
        tdm_load_tile16x32(lds0, gbase);

    for (int ci = 0; ci < NCH; ci++) {
        if (wave == 0) {
            if (ci + 1 < NCH) {               // issue next chunk, then wait
                tdm_load_tile16x32((ci + 1) & 1 ? lds1 : lds0,
                                   gbase + (size_t)(ci + 1) * CH * 96);
                __builtin_amdgcn_s_wait_tensorcnt(1);
            } else {
                __builtin_amdgcn_s_wait_tensorcnt(0);
            }
        }
        __syncthreads();                      // publish chunk ci's LDS tile
        const float (*buf)[32] = lBC[ci & 1];
#pragma unroll 4
        for (int ls = 0; ls < CH; ++ls) {
            int l = ci * CH + ls;
            size_t base = ((size_t)(b * LL) + l) * D_INNER + d;
            float dtv = dtb[base];
            float xv  = xc[base];
            float zv  = xz[((size_t)(b * LL) + l) * (2 * D_INNER) + D_INNER + d];
            float y = 0.0f;
#pragma unroll
            for (int n = 0; n < D_STATE; n++) {
                float dA = __expf(dtv * Ad[n]);
                s[n] = dA * s[n] + (dtv * buf[ls][n]) * xv;
                y += s[n] * buf[ls][16 + n];
            }
            y += Dd * xv;
            y *= zv / (1.0f + __expf(-zv));   // * silu(z)
            xc[base] = y;
        }
        __syncthreads();                      // done reading buf[ci&1] before
    }                                         // it is overwritten at ci+2
}

// ---------------------------------------------------------------------------
extern "C" void kernel_launch(void* const* d_in, const int* in_sizes, int n_in,
                              void* d_out, int out_size, void* d_ws, size_t ws_size,
                              hipStream_t stream)
{
    const float* hs     = (const float*)d_in[0];   // hidden_states [2,1024,1024]
    const float* scale  = (const float*)d_in[1];   // [1024]
    const float* W_in   = (const float*)d_in[2];   // [4096,1024]
    const float* conv_w = (const float*)d_in[3];   // [2048,1,4]
    const float* conv_b = (const float*)d_in[4];   // [2048]
    const float* W_x    = (const float*)d_in[5];   // [96,2048]
    const float* W_dt   = (const float*)d_in[6];   // [2048,64]
    const float* b_dt   = (const float*)d_in[7];   // [2048]
    const float* A_log  = (const float*)d_in[8];   // [2048,16]
    const float* Dp     = (const float*)d_in[9];   // [2048]
    const float* W_out  = (const float*)d_in[10];  // [1024,2048]
    float* out = (float*)d_out;                    // [2,1024,1024]

    // workspace layout (floats): ~76 MB total
    float* ws   = (float*)d_ws;
    float* hn   = ws;                                 // NTOK*D_MODEL   = 2,097,152
    float* xz   = hn   + (size_t)NTOK * D_MODEL;      // NTOK*2*D_INNER = 8,388,608
    float* xc   = xz   + (size_t)NTOK * 2 * D_INNER;  // NTOK*D_INNER   = 4,194,304
    float* xdbl = xc   + (size_t)NTOK * D_INNER;      // NTOK*96        =   196,608
    float* dtb  = xdbl + (size_t)NTOK * 96;           // NTOK*D_INNER   = 4,194,304

    // 1) RMSNorm
    rmsnorm_kernel<<<NTOK, 256, 0, stream>>>(hs, scale, hn);

    // 2) in_proj: [2048,1024] x [4096,1024]^T -> xz [2048,4096]
    gemm_wmma_f32<8, 2, 0><<<dim3(NTOK / 64, (2 * D_INNER) / 256), 256, 0, stream>>>(
        hn, D_MODEL, W_in, D_MODEL, xz, 2 * D_INNER, nullptr, 0);

    // 3) causal depthwise conv + SiLU on x half
    conv_silu_kernel<<<(NTOK * D_INNER) / 256, 256, 0, stream>>>(
        xz, conv_w, conv_b, xc);

    // 4) x_proj: [2048,2048] x [96,2048]^T -> x_dbl [2048,96]
    //    NT=6, WN=1: block tile 128M x 96N -> exact N coverage, no partials.
    gemm_wmma_f32<6, 1, 0><<<dim3(NTOK / 128, 1), 256, 0, stream>>>(
        xc, D_INNER, W_x, D_INNER, xdbl, 96, nullptr, 0);

    // 5) dt_proj + softplus: [2048,64(ld 96)] x [2048,64]^T -> dt [2048,2048]
    gemm_wmma_f32<8, 2, 1><<<dim3(NTOK / 64, D_INNER / 256), 256, 0, stream>>>(
        xdbl, 96, W_dt, DT_RANK, dtb, D_INNER, b_dt, 0);

    // 6) selective scan (TDM-staged B/C, fuses +D*x and *silu(z))
    scan_kernel<<<BB * (D_INNER / 256), 256, 0, stream>>>(
        xdbl, dtb, xz, xc, A_log, Dp);

    // 7) out_proj + residual: [2048,2048] x [1024,2048]^T + hs -> out
    gemm_wmma_f32<8, 2, 2><<<dim3(NTOK / 64, D_MODEL / 256), 256, 0, stream>>>(
        xc, D_INNER, W_out, D_INNER, out, D_MODEL, hs, D_MODEL);
}